// StreamSpatialGCN_1881195675774
// MI455X (gfx1250) — compile-verified
//
#include <hip/hip_runtime.h>
#include <cstddef>

// ---------------------------------------------------------------------------
// StreamSpatialGCN (2s-AGCN style) forward for MI455X / gfx1250.
//
// Bandwidth-bound (~40 GFLOP, ~2.5 GB of f32 activation traffic -> ~110us
// floor at 23.3 TB/s; 123MB tensors live in the 192MB L2).  Kept in f32 to
// match reference numerics; the two GEMM-shaped stages (25x(inter*T)x25
// attention Gram and (C*T)x25x25 graph mixing) use the CDNA5 FP32 WMMA
// V_WMMA_F32_16X16X4_F32 via __builtin_amdgcn_wmma_f32_16x16x4_f32.
//
// WMMA fragment mapping (per CDNA5 ISA, 32-bit 16x4 A / 4x16 B):
//   A: lane l holds A[m][kb], A[m][kb+1] with m = l&15, kb = 2*(l>>4)
//   B: lane l holds B[kb][n], B[kb+1][n] with n = l&15, kb = 2*(l>>4)
//   C/D: VGPR r, lane l -> row = r + 8*(l>>4), col = l&15
// Out-of-range M/N rows/cols are handled by address clamping (their products
// land in D rows/cols we never store), so the inner loops have no divergent
// loads / EXEC manipulation; only the padded K tail in the vmix kernel zeroes
// the B fragment (constant-folded by full unroll).
// ---------------------------------------------------------------------------

typedef float v2f __attribute__((ext_vector_type(2)));
typedef float v8f __attribute__((ext_vector_type(8)));

constexpr int NV  = 25;      // joints
constexpr int NT  = 300;     // frames
constexpr int NTV = NV * NT; // 7500
constexpr int NS  = 256;     // N*M samples
constexpr float BNEPS = 1e-5f;

static __device__ __forceinline__ float bn1(float x, float g, float b, float m, float v) {
  return (x - m) * g * rsqrtf(v + BNEPS) + b;
}

// ---- data BN + permute: x(N,3,T,V,M) -> h(N*M,3,T,V) ----------------------
__global__ void k_databn(const float* __restrict__ x,
                         const float* __restrict__ g, const float* __restrict__ b,
                         const float* __restrict__ mu, const float* __restrict__ var,
                         float* __restrict__ h) {
  int idx = blockIdx.x * blockDim.x + threadIdx.x;
  if (idx >= NS * 3 * NTV) return;
  int vv = idx % NV; int r = idx / NV;
  int t  = r % NT;   r /= NT;
  int c  = r % 3;    r /= 3;
  int s  = r;                       // n*2+m
  int m  = s & 1, n = s >> 1;
  int xi = (((n * 3 + c) * NT + t) * NV + vv) * 2 + m;
  int ch = m * NV * 3 + vv * 3 + c; // (m, v, c) channel of the 150-ch data BN
  h[idx] = bn1(x[xi], g[ch], b[ch], mu[ch], var[ch]);
}

// ---- pointwise channel mix (conv 1x1), optional accumulate ----------------
template <int CIN, int COUT>
__global__ void k_conv1x1(const float* __restrict__ src, const float* __restrict__ w,
                          const float* __restrict__ bias, float* __restrict__ dst,
                          int accum) {
  int idx = blockIdx.x * blockDim.x + threadIdx.x;   // (s, tv)
  if (idx >= NS * NTV) return;
  int s = idx / NTV, tv = idx % NTV;
  const float* sp = src + (size_t)s * CIN * NTV + tv;
  float xr[CIN];
#pragma unroll
  for (int c = 0; c < CIN; ++c) xr[c] = sp[(size_t)c * NTV];
  float* dp = dst + (size_t)s * COUT * NTV + tv;
#pragma unroll
  for (int o = 0; o < COUT; ++o) {
    float acc = bias[o];
#pragma unroll
    for (int c = 0; c < CIN; ++c) acc += w[o * CIN + c] * xr[c];
    if (accum) dp[(size_t)o * NTV] += acc;
    else       dp[(size_t)o * NTV]  = acc;
  }
}

// ---- S[s,u,v] = scale * sum_k a[s,k,u]*b[s,k,v]  (K = inter*T) ------------
// one 128-thread block (4 waves) per sample; each wave owns a 16x16 tile of
// the zero-padded 32x32 output.  f32 WMMA, K consumed 4 at a time.
// Rows/cols >= 25 use clamped addresses; their garbage lands only in D
// rows/cols that are never stored, so no masks and no divergent loads.
__global__ void k_gram_wmma(const float* __restrict__ a, const float* __restrict__ bm,
                            float* __restrict__ S, int K, float scale) {
  int s    = blockIdx.x;
  int wave = threadIdx.x >> 5;
  int lane = threadIdx.x & 31;
  int tm = wave >> 1, tn = wave & 1;
  const float* ap = a  + (size_t)s * K * NV;
  const float* bp = bm + (size_t)s * K * NV;
  int mrow = tm * 16 + (lane & 15);
  int ncol = tn * 16 + (lane & 15);
  int mcl  = mrow < NV ? mrow : NV - 1;    // address clamp only
  int ncl  = ncol < NV ? ncol : NV - 1;
  int koff = (lane >> 4) << 1;
  const float* apk = ap + (size_t)koff * NV + mcl;   // &A1[mcl][koff]
  const float* bpk = bp + (size_t)koff * NV + ncl;   // &A2[koff][ncl]
  v8f acc = {};
  for (int k0 = 0; k0 < K; k0 += 4) {      // K is a multiple of 4 (600/1200)
    v2f A, B;
    A.x = apk[0];  A.y = apk[NV];
    B.x = bpk[0];  B.y = bpk[NV];
    acc = __builtin_amdgcn_wmma_f32_16x16x4_f32(false, A, false, B,
                                                (short)0, acc, false, false);
    apk += 4 * NV; bpk += 4 * NV;
  }
  float* Sp = S + (size_t)s * NV * NV;
#pragma unroll
  for (int r = 0; r < 8; ++r) {
    int row = tm * 16 + r + ((lane >> 4) << 3);
    int col = tn * 16 + (lane & 15);
    if (row < NV && col < NV) Sp[row * NV + col] = acc[r] * scale;
  }
}

// ---- column softmax over rows (torch Softmax(-2)) + A[i] + 1e-6 -----------
__global__ void k_softmax_bc(const float* __restrict__ S, const float* __restrict__ Ai,
                             float* __restrict__ BC) {
  int idx = blockIdx.x * blockDim.x + threadIdx.x;  // (s, col)
  if (idx >= NS * NV) return;
  int s = idx / NV, col = idx % NV;
  const float* Sp = S + (size_t)s * NV * NV + col;
  float mx = -1e30f;
#pragma unroll
  for (int u = 0; u < NV; ++u) mx = fmaxf(mx, Sp[u * NV]);
  float e[NV]; float sum = 0.f;
#pragma unroll
  for (int u = 0; u < NV; ++u) { e[u] = __expf(Sp[u * NV] - mx); sum += e[u]; }
  float inv = 1.f / sum;
  float* Bp = BC + (size_t)s * NV * NV + col;
#pragma unroll
  for (int u = 0; u < NV; ++u) Bp[u * NV] = e[u] * inv + Ai[u * NV + col] + 1e-6f;
}

// ---- Z[s,r,v] = sum_u X[s,r,u]*BC[s,u,v],  r in [0,C*T) -------------------
// one wave per (sample, 16-row tile, 16-col tile); K=25 padded to 28.
// M/N handled by address clamping; only the K tail zeroes the B fragment
// (A-side garbage * 0 contributes nothing), and those masks constant-fold.
__global__ void k_vmix_wmma(const float* __restrict__ X, const float* __restrict__ BC,
                            float* __restrict__ Z, int CT, int tilesM) {
  int gw   = (blockIdx.x * blockDim.x + threadIdx.x) >> 5;
  int lane = threadIdx.x & 31;
  int totalW = NS * tilesM * 2;
  if (gw >= totalW) return;                 // uniform per-wave exit
  int tn = gw & 1; int r0 = gw >> 1;
  int tm = r0 % tilesM; int s = r0 / tilesM;
  const float* xp = X  + (size_t)s * CT * NV;
  const float* bp = BC + (size_t)s * NV * NV;
  int mrow = tm * 16 + (lane & 15);
  int ncol = tn * 16 + (lane & 15);
  int mcl  = mrow < CT ? mrow : CT - 1;     // address clamp
  int ncl  = ncol < NV ? ncol : NV - 1;
  int koff = (lane >> 4) << 1;
  const float* xr = xp + (size_t)mcl * NV;  // row base of X
  v8f acc = {};
#pragma unroll
  for (int k0 = 0; k0 < 28; k0 += 4) {
    int kb  = k0 + koff;
    int kb0 = kb     < NV ? kb     : NV - 1;
    int kb1 = kb + 1 < NV ? kb + 1 : NV - 1;
    float km0 = kb     < NV ? 1.f : 0.f;    // constant-folds except last step
    float km1 = kb + 1 < NV ? 1.f : 0.f;
    v2f A, B;
    A.x = xr[kb0];
    A.y = xr[kb1];
    B.x = bp[kb0 * NV + ncl] * km0;
    B.y = bp[kb1 * NV + ncl] * km1;
    acc = __builtin_amdgcn_wmma_f32_16x16x4_f32(false, A, false, B,
                                                (short)0, acc, false, false);
  }
  float* zp = Z + (size_t)s * CT * NV;
#pragma unroll
  for (int r = 0; r < 8; ++r) {
    int row = tm * 16 + r + ((lane >> 4) << 3);
    int col = tn * 16 + (lane & 15);
    if (row < CT && col < NV) zp[(size_t)row * NV + col] = acc[r];
  }
}

// ---- AGC epilogue: BN(fusion) + residual (conv+BN or identity), ReLU ------
template <int CIN, int COUT>
__global__ void k_agc_epi(const float* __restrict__ fus, const float* __restrict__ hin,
                          const float* __restrict__ bg, const float* __restrict__ bb,
                          const float* __restrict__ bm, const float* __restrict__ bv,
                          const float* __restrict__ rw, const float* __restrict__ rb,
                          const float* __restrict__ rg, const float* __restrict__ rbb,
                          const float* __restrict__ rm, const float* __restrict__ rv,
                          float* __restrict__ out) {
  int idx = blockIdx.x * blockDim.x + threadIdx.x;
  if (idx >= NS * COUT * NTV) return;
  int tv = idx % NTV; int r = idx / NTV;
  int o = r % COUT; int s = r / COUT;
  float f = bn1(fus[idx], bg[o], bb[o], bm[o], bv[o]);
  float res;
  if (rw) {
    float acc = rb[o];
    const float* hp = hin + (size_t)s * CIN * NTV + tv;
#pragma unroll
    for (int c = 0; c < CIN; ++c) acc += rw[o * CIN + c] * hp[(size_t)c * NTV];
    res = bn1(acc, rg[o], rbb[o], rm[o], rv[o]);
  } else {
    res = hin[((size_t)s * CIN + o) * NTV + tv];  // CIN==COUT here
  }
  out[idx] = fmaxf(f + res, 0.f);
}

// ---- temporal conv (9,1) pad 4 + BN + block residual + ReLU ---------------
template <int CIN, int COUT>
__global__ void k_tconv(const float* __restrict__ y, const float* __restrict__ tw,
                        const float* __restrict__ tb,
                        const float* __restrict__ tg, const float* __restrict__ tbeta,
                        const float* __restrict__ tmean, const float* __restrict__ tvar,
                        const float* __restrict__ hin,
                        const float* __restrict__ brw, const float* __restrict__ brb,
                        const float* __restrict__ rg, const float* __restrict__ rbb,
                        const float* __restrict__ rm, const float* __restrict__ rv,
                        float* __restrict__ out, int resmode) {  // 0 none,1 id,2 conv
  int idx = blockIdx.x * blockDim.x + threadIdx.x;  // (s, t, v)
  if (idx >= NS * NTV) return;
  int s = idx / NTV; int tv = idx % NTV;
  int t = tv / NV;   int vv = tv % NV;
  float accs[COUT];
#pragma unroll
  for (int o = 0; o < COUT; ++o) accs[o] = tb[o];
  const float* yp = y + (size_t)s * COUT * NTV;
#pragma unroll
  for (int k = 0; k < 9; ++k) {
    int tt = t + k - 4;
    if (tt < 0 || tt >= NT) continue;
#pragma unroll
    for (int c = 0; c < COUT; ++c) {
      float xv = yp[(size_t)c * NTV + tt * NV + vv];
#pragma unroll
      for (int o = 0; o < COUT; ++o) accs[o] += tw[(o * COUT + c) * 9 + k] * xv;
    }
  }
#pragma unroll
  for (int o = 0; o < COUT; ++o) {
    float val = bn1(accs[o], tg[o], tbeta[o], tmean[o], tvar[o]);
    float r = 0.f;
    if (resmode == 1) {
      r = hin[((size_t)s * CIN + o) * NTV + tv];
    } else if (resmode == 2) {
      float acc = brb[o];
#pragma unroll
      for (int c = 0; c < CIN; ++c)
        acc += brw[o * CIN + c] * hin[((size_t)s * CIN + c) * NTV + tv];
      r = bn1(acc, rg[o], rbb[o], rm[o], rv[o]);
    }
    out[((size_t)s * COUT + o) * NTV + tv] = fmaxf(val + r, 0.f);
  }
}

// ---- h(N*M,16,T,V) -> out(N,16,T,V,M) -------------------------------------
__global__ void k_output(const float* __restrict__ h, float* __restrict__ out) {
  int idx = blockIdx.x * blockDim.x + threadIdx.x;
  int total = 128 * 16 * NTV * 2;
  if (idx >= total) return;
  int m  = idx % 2;  int r = idx / 2;
  int vv = r % NV;   r /= NV;
  int t  = r % NT;   r /= NT;
  int co = r % 16;   r /= 16;
  int n  = r;
  out[idx] = h[(((size_t)(n * 2 + m) * 16 + co) * NT + t) * NV + vv];
}

// ---------------------------------------------------------------------------
// host side
// ---------------------------------------------------------------------------
static void conv1x1(const float* s, const float* w, const float* b, float* d,
                    int ci, int co, int acc, hipStream_t st) {
  int g = (NS * NTV + 255) / 256;
#define CASE(CI, CO) if (ci == CI && co == CO) { k_conv1x1<CI, CO><<<g, 256, 0, st>>>(s, w, b, d, acc); return; }
  CASE(3, 2) CASE(8, 2) CASE(8, 4) CASE(16, 4)
  CASE(3, 8) CASE(8, 8) CASE(8, 16) CASE(16, 16)
#undef CASE
}

struct LayerP {
  const float *wa, *ba, *wb, *bb, *wd, *bd, *A, *bn[4];
  const float *rw, *rb, *rbn[4];
  const float *tw, *tb, *tbn[4];
  const float *brw, *brb, *brbn[4];
};

extern "C" void kernel_launch(void* const* d_in, const int* in_sizes, int n_in,
                              void* d_out, int out_size, void* d_ws, size_t ws_size,
                              hipStream_t stream) {
  (void)in_sizes; (void)n_in; (void)out_size; (void)ws_size;
  int p = 0;
  auto nxt = [&]() { return (const float*)d_in[p++]; };

  const float* x = nxt();
  const float* dbn[4]; for (int i = 0; i < 4; ++i) dbn[i] = nxt();

  const int cins[7]  = {3, 8, 8, 8, 8, 16, 16};
  const int couts[7] = {8, 8, 8, 8, 16, 16, 16};
  LayerP L[7];
  for (int i = 0; i < 7; ++i) {
    bool res = cins[i] != couts[i];
    LayerP& l = L[i];
    l.wa = nxt(); l.ba = nxt(); l.wb = nxt(); l.bb = nxt();
    l.wd = nxt(); l.bd = nxt(); l.A  = nxt();
    for (int j = 0; j < 4; ++j) l.bn[j] = nxt();
    if (res) { l.rw = nxt(); l.rb = nxt(); for (int j = 0; j < 4; ++j) l.rbn[j] = nxt(); }
    else     { l.rw = nullptr; l.rb = nullptr; for (int j = 0; j < 4; ++j) l.rbn[j] = nullptr; }
    l.tw = nxt(); l.tb = nxt();
    for (int j = 0; j < 4; ++j) l.tbn[j] = nxt();
    if (res) { l.brw = nxt(); l.brb = nxt(); for (int j = 0; j < 4; ++j) l.brbn[j] = nxt(); }
    else     { l.brw = nullptr; l.brb = nullptr; for (int j = 0; j < 4; ++j) l.brbn[j] = nullptr; }
  }

  // workspace layout
  char* ws = (char*)d_ws;
  size_t BIG = (size_t)NS * 16 * NTV * sizeof(float);  // 122.88 MB
  size_t SM  = (size_t)NS * 4  * NTV * sizeof(float);  // 30.72 MB
  size_t ATT = (size_t)NS * NV * NV * sizeof(float);   // 0.64 MB
  float* buf0 = (float*)ws;                  // ping activations
  float* buf1 = (float*)(ws + BIG);          // pong activations / z scratch
  float* fus  = (float*)(ws + 2 * BIG);      // fusion accumulator
  float* abuf = (float*)(ws + 3 * BIG);
  float* bbuf = (float*)(ws + 3 * BIG + SM);
  float* Sbuf = (float*)(ws + 3 * BIG + 2 * SM);
  float* BCb  = (float*)(ws + 3 * BIG + 2 * SM + ATT);

  // data BN + permute
  {
    int tot = NS * 3 * NTV;
    k_databn<<<(tot + 255) / 256, 256, 0, stream>>>(x, dbn[0], dbn[1], dbn[2], dbn[3], buf0);
  }

  float* hin = buf0; float* hout = buf1;
  for (int i = 0; i < 7; ++i) {
    int cin = cins[i], cout = couts[i], inter = cout / 4;
    int K = inter * NT;
    int CT = cin * NT;
    int tilesM = (CT + 15) / 16;

    for (int sub = 0; sub < 3; ++sub) {
      conv1x1(hin, L[i].wa + sub * inter * cin, L[i].ba + sub * inter, abuf, cin, inter, 0, stream);
      conv1x1(hin, L[i].wb + sub * inter * cin, L[i].bb + sub * inter, bbuf, cin, inter, 0, stream);
      k_gram_wmma<<<NS, 128, 0, stream>>>(abuf, bbuf, Sbuf, K, 1.0f / (float)K);
      k_softmax_bc<<<(NS * NV + 255) / 256, 256, 0, stream>>>(Sbuf, L[i].A + sub * NV * NV, BCb);
      int totW = NS * tilesM * 2;
      k_vmix_wmma<<<(totW + 3) / 4, 128, 0, stream>>>(hin, BCb, hout, CT, tilesM);
      conv1x1(hout, L[i].wd + sub * cout * cin, L[i].bd + sub * cout, fus, cin, cout, sub > 0, stream);
    }

    // AGC epilogue (in place on fus)
    {
      int tot = NS * cout * NTV;
      int g = (tot + 255) / 256;
#define EPI(CI, CO) if (cin == CI && cout == CO) \
      k_agc_epi<CI, CO><<<g, 256, 0, stream>>>(fus, hin, \
          L[i].bn[0], L[i].bn[1], L[i].bn[2], L[i].bn[3], \
          L[i].rw, L[i].rb, L[i].rbn[0], L[i].rbn[1], L[i].rbn[2], L[i].rbn[3], fus);
      EPI(3, 8) EPI(8, 8) EPI(8, 16) EPI(16, 16)
#undef EPI
    }

    // temporal conv + BN + block residual + relu -> hout
    {
      int resmode = (i == 0) ? 0 : (L[i].brw ? 2 : 1);
      int g = (NS * NTV + 255) / 256;
#define TCV(CI, CO) if (cin == CI && cout == CO) \
      k_tconv<CI, CO><<<g, 256, 0, stream>>>(fus, L[i].tw, L[i].tb, \
          L[i].tbn[0], L[i].tbn[1], L[i].tbn[2], L[i].tbn[3], \
          hin, L[i].brw, L[i].brb, \
          L[i].brbn[0], L[i].brbn[1], L[i].brbn[2], L[i].brbn[3], hout, resmode);
      TCV(3, 8) TCV(8, 8) TCV(8, 16) TCV(16, 16)
#undef TCV
    }

    float* t = hin; hin = hout; hout = t;
  }

  // final permute to (N, 16, T, V, M)
  {
    int tot = 128 * 16 * NTV * 2;
    k_output<<<(tot + 255) / 256, 256, 0, stream>>>(hin, (float*)d_out);
  }
}